// Scatter_layer_84851373900280
// MI455X (gfx1250) — compile-verified
//
#include <hip/hip_runtime.h>
#include <math.h>

// Problem constants (fixed by the reference).
#define N_NODES  50000
#define N_EDGES  800000
#define C_DIM    64
#define N_LAYERS 4
#define K_STEPS  16
#define N_TILES  (N_NODES / 16)   // 3125 row tiles of 16

typedef __attribute__((ext_vector_type(2))) float v2f;
typedef __attribute__((ext_vector_type(8))) float v8f;

// ---------------------------------------------------------------------------
// WMMA helper: one wave computes a 16-row x 64-col fp32 tile of  A[16x64] @ W[64x64]
// using V_WMMA_F32_16X16X4_F32.  A-fragment layout (16x4 MxK, 32-bit):
//   lanes 0-15 = M, VGPR0 holds K = 0 (lo half) / K = 2 (hi half), VGPR1 K = 1 / 3.
// B-fragment (4x16 KxN): lanes carry N, same K striping.
// C/D: VGPR i -> row i + 8*(lane>=16), column = lane&15 (+16*t per tile).
// ---------------------------------------------------------------------------
__device__ __forceinline__ void wave_gemm_16x64(const float* __restrict__ A,
                                                const float* __restrict__ W,
                                                v8f acc[4], int lane) {
  const int ph = lane >> 4;   // half of wave
  const int r  = lane & 15;
#pragma unroll
  for (int kk = 0; kk < 64; kk += 4) {
    v2f a;
    a.x = A[r * 64 + kk + 2 * ph + 0];
    a.y = A[r * 64 + kk + 2 * ph + 1];
#pragma unroll
    for (int t = 0; t < 4; ++t) {
      v2f b;
      b.x = W[(kk + 2 * ph + 0) * 64 + t * 16 + r];
      b.y = W[(kk + 2 * ph + 1) * 64 + t * 16 + r];
      acc[t] = __builtin_amdgcn_wmma_f32_16x16x4_f32(
          false, a, false, b, (short)0, acc[t], false, false);
    }
  }
}

// ---------------------------------------------------------------------------
// Small elementwise / graph kernels
// ---------------------------------------------------------------------------
__global__ void zero_f32(float* __restrict__ p, int n) {
  int i = blockIdx.x * blockDim.x + threadIdx.x;
  if (i < n) p[i] = 0.f;
}

__global__ void deg_count(const int* __restrict__ col, float* __restrict__ deg) {
  int e = blockIdx.x * blockDim.x + threadIdx.x;
  if (e < N_EDGES) atomicAdd(&deg[col[e]], 1.0f);
}

__global__ void edge_norm(const int* __restrict__ row, const float* __restrict__ deg,
                          float* __restrict__ norm) {
  int e = blockIdx.x * blockDim.x + threadIdx.x;
  if (e < N_EDGES) {
    float d = deg[row[e]];
    norm[e] = (d > 0.f) ? (1.0f / d) : 0.0f;
  }
}

// v = h ; psi = w0 * h ; agg = 0   (fused so the step loop needs no zero kernel)
__global__ void psi_init_kernel(const float* __restrict__ h, float* __restrict__ v,
                                float* __restrict__ psi, float* __restrict__ agg,
                                const float* __restrict__ wavelet, int l) {
  int i = blockIdx.x * blockDim.x + threadIdx.x;
  if (i < N_NODES * C_DIM) {
    float w0 = wavelet[l * (K_STEPS + 1)] -
               (l ? wavelet[(l - 1) * (K_STEPS + 1)] : 0.f);
    float hv = h[i];
    v[i]   = hv;
    psi[i] = w0 * hv;
    agg[i] = 0.f;
  }
}

// Scatter: agg[col[e], :] += norm[e] * v[row[e], :]  (one thread per edge x 4 channels)
__global__ void scatter_kernel(const int* __restrict__ row, const int* __restrict__ col,
                               const float* __restrict__ norm,
                               const float* __restrict__ v, float* __restrict__ agg) {
  int tid = blockIdx.x * blockDim.x + threadIdx.x;
  int e = tid >> 4;
  if (e >= N_EDGES) return;
  int c4 = (tid & 15) << 2;
  int rs = row[e];
  int cd = col[e];
  float nrm = norm[e];
  const float4 vv = *reinterpret_cast<const float4*>(v + (size_t)rs * C_DIM + c4);
  float* dst = agg + (size_t)cd * C_DIM + c4;
  atomicAdd(dst + 0, nrm * vv.x);
  atomicAdd(dst + 1, nrm * vv.y);
  atomicAdd(dst + 2, nrm * vv.z);
  atomicAdd(dst + 3, nrm * vv.w);
}

// v = 0.5*(v+agg) ; psi += wk * v ; agg = 0 (ready for next step)
__global__ void combine_kernel(float* __restrict__ v, float* __restrict__ agg,
                               float* __restrict__ psi,
                               const float* __restrict__ wavelet, int l, int k) {
  int i = blockIdx.x * blockDim.x + threadIdx.x;
  if (i < N_NODES * C_DIM) {
    float wk = wavelet[l * (K_STEPS + 1) + k] -
               (l ? wavelet[(l - 1) * (K_STEPS + 1) + k] : 0.f);
    float vn = 0.5f * (v[i] + agg[i]);
    v[i] = vn;
    psi[i] += wk * vn;
    agg[i] = 0.f;
  }
}

// ---------------------------------------------------------------------------
// h = x @ W_in   (WMMA, no epilogue)
// ---------------------------------------------------------------------------
__global__ void gemm_in_kernel(const float* __restrict__ x, const float* __restrict__ Win,
                               float* __restrict__ h) {
  const int lane = threadIdx.x & 31;
  const int wave = threadIdx.x >> 5;
  const int tile = blockIdx.x * 4 + wave;
  if (tile >= N_TILES) return;
  const int m0 = tile * 16;
  const int ph = lane >> 4, r = lane & 15;

  v8f zero = {};
  v8f acc[4] = {zero, zero, zero, zero};
  wave_gemm_16x64(x + (size_t)m0 * 64, Win, acc, lane);

#pragma unroll
  for (int t = 0; t < 4; ++t)
#pragma unroll
    for (int i = 0; i < 8; ++i)
      h[(size_t)(m0 + i + 8 * ph) * 64 + t * 16 + r] = acc[t][i];
}

// ---------------------------------------------------------------------------
// Fused dense block per layer:
//   y = psi @ W1[l] + b1[l];  LayerNorm(gamma,beta);  gelu(exact);
//   h = y @ W2[l] + b2[l];    out[n, l*64+c] = h[n,c]
// One wave per 16-row tile; LDS restage of y between the two WMMA GEMMs.
// ---------------------------------------------------------------------------
__global__ void dense_block_kernel(const float* __restrict__ psi,
                                   const float* __restrict__ W1, const float* __restrict__ b1,
                                   const float* __restrict__ gmm, const float* __restrict__ bta,
                                   const float* __restrict__ W2, const float* __restrict__ b2,
                                   float* __restrict__ h, float* __restrict__ out, int l) {
  __shared__ float ybuf[4 * 16 * 64];
  const int lane = threadIdx.x & 31;
  const int wave = threadIdx.x >> 5;
  const int tile = blockIdx.x * 4 + wave;
  if (tile >= N_TILES) return;   // wave-uniform: EXEC stays all-ones for WMMA
  const int m0 = tile * 16;
  const int ph = lane >> 4, r = lane & 15;
  float* y = ybuf + wave * 16 * 64;

  const float* W1l = W1 + (size_t)l * 64 * 64;
  const float* W2l = W2 + (size_t)l * 64 * 64;

  v8f zero = {};
  v8f acc[4] = {zero, zero, zero, zero};
  wave_gemm_16x64(psi + (size_t)m0 * 64, W1l, acc, lane);

  // per-column params for this lane's 4 columns
  float bc[4], gc[4], btc[4], b2c[4];
#pragma unroll
  for (int t = 0; t < 4; ++t) {
    int c = t * 16 + r;
    bc[t]  = b1[l * 64 + c];
    gc[t]  = gmm[l * 64 + c];
    btc[t] = bta[l * 64 + c];
    b2c[t] = b2[l * 64 + c];
  }

  // bias
#pragma unroll
  for (int t = 0; t < 4; ++t)
#pragma unroll
    for (int i = 0; i < 8; ++i) acc[t][i] += bc[t];

  // LayerNorm (row stats via in-register sum over t + butterfly over the 16-lane
  // half that owns the row) then exact gelu; stage to LDS in A-operand order.
#pragma unroll
  for (int i = 0; i < 8; ++i) {
    float s = 0.f, s2 = 0.f;
#pragma unroll
    for (int t = 0; t < 4; ++t) { float vv = acc[t][i]; s += vv; s2 += vv * vv; }
#pragma unroll
    for (int m = 1; m <= 8; m <<= 1) {
      s  += __shfl_xor(s,  m, 32);
      s2 += __shfl_xor(s2, m, 32);
    }
    float mu   = s * (1.0f / 64.0f);
    float var  = s2 * (1.0f / 64.0f) - mu * mu;
    float rstd = rsqrtf(var + 1e-5f);
#pragma unroll
    for (int t = 0; t < 4; ++t) {
      float yv = (acc[t][i] - mu) * rstd * gc[t] + btc[t];
      float g  = 0.5f * yv * (1.0f + erff(yv * 0.70710678118654752f));
      y[(i + 8 * ph) * 64 + t * 16 + r] = g;
    }
  }

  // wave-local LDS RAW fence before re-reading y as the A operand
  asm volatile("s_wait_dscnt 0" ::: "memory");

  v8f acc2[4] = {zero, zero, zero, zero};
  wave_gemm_16x64(y, W2l, acc2, lane);

#pragma unroll
  for (int t = 0; t < 4; ++t) {
    int c = t * 16 + r;
#pragma unroll
    for (int i = 0; i < 8; ++i) {
      int rowi = m0 + i + 8 * ph;
      float hv = acc2[t][i] + b2c[t];
      h[(size_t)rowi * 64 + c] = hv;
      out[(size_t)rowi * (N_LAYERS * 64) + l * 64 + c] = hv;   // dense-batch reshape is identity
    }
  }
}

// ---------------------------------------------------------------------------
extern "C" void kernel_launch(void* const* d_in, const int* in_sizes, int n_in,
                              void* d_out, int out_size, void* d_ws, size_t ws_size,
                              hipStream_t stream) {
  const float* x       = (const float*)d_in[0];
  const int*   edge    = (const int*)d_in[1];
  const int*   row     = edge;             // edge_index[0]
  const int*   col     = edge + N_EDGES;   // edge_index[1]
  // d_in[2] batch, d_in[3] num_graphs: unused (reshape is identity for this layout)
  const float* Win     = (const float*)d_in[4];
  const float* wavelet = (const float*)d_in[5];
  const float* W1      = (const float*)d_in[6];
  const float* b1      = (const float*)d_in[7];
  const float* gmm     = (const float*)d_in[8];
  const float* bta     = (const float*)d_in[9];
  const float* W2      = (const float*)d_in[10];
  const float* b2      = (const float*)d_in[11];
  float*       out     = (float*)d_out;

  // workspace layout (floats)
  float* ws   = (float*)d_ws;
  float* deg  = ws;                               // N
  float* norm = ws + 51200;                       // E (aligned past deg)
  float* h    = norm + N_EDGES;                   // N*64
  float* v    = h   + (size_t)N_NODES * C_DIM;    // N*64
  float* agg  = v   + (size_t)N_NODES * C_DIM;    // N*64
  float* psi  = agg + (size_t)N_NODES * C_DIM;    // N*64

  const int total      = N_NODES * C_DIM;
  const int gemmBlocks = (N_TILES + 3) / 4;       // 4 waves (tiles) per 128-thread block

  // degree / edge normalization
  zero_f32 <<<(N_NODES + 255) / 256, 256, 0, stream>>>(deg, N_NODES);
  deg_count<<<(N_EDGES + 255) / 256, 256, 0, stream>>>(col, deg);
  edge_norm<<<(N_EDGES + 255) / 256, 256, 0, stream>>>(row, deg, norm);

  // h = x @ W_in
  gemm_in_kernel<<<gemmBlocks, 128, 0, stream>>>(x, Win, h);

  for (int l = 0; l < N_LAYERS; ++l) {
    psi_init_kernel<<<(total + 255) / 256, 256, 0, stream>>>(h, v, psi, agg, wavelet, l);
    for (int k = 1; k <= K_STEPS; ++k) {
      scatter_kernel<<<(N_EDGES * 16 + 255) / 256, 256, 0, stream>>>(row, col, norm, v, agg);
      combine_kernel<<<(total + 255) / 256, 256, 0, stream>>>(v, agg, psi, wavelet, l, k);
    }
    dense_block_kernel<<<gemmBlocks, 128, 0, stream>>>(psi, W1, b1, gmm, bta, W2, b2,
                                                       h, out, l);
  }
}